// biedgeGraphConvolution_32238024524116
// MI455X (gfx1250) — compile-verified
//
#include <hip/hip_runtime.h>
#include <hip/hip_bf16.h>

typedef __attribute__((ext_vector_type(2))) float v2f;
typedef __attribute__((ext_vector_type(4))) float f32x4;
typedef __attribute__((ext_vector_type(8))) float v8f;

#define WAVES_PER_BLOCK 4

// ---------------------------------------------------------------------------
// WMMA f32 GEMM:  C[M,Nout] = A[M,K] * Bsw[K,Nout]   (A row-major, B swizzled)
//
// B is pre-swizzled so element (k,n) lives at ((k>>2)*Nout + n)*4 + (k&3):
// a lane's 4x16 B fragment (two K values at one column) is one contiguous
// 8-byte load. Each 64-deep K chunk of the swizzled B is a contiguous
// 64*Nout*4-byte block, staged cooperatively into LDS and shared by all 4
// waves of the block (4x less L2 traffic, B frags become ds_load_2addr_b64).
// A fragments for a whole chunk are batch-preloaded (16 global_load_b64) and
// the k-step loop is fully unrolled so af[] indexing is static (no MOVREL).
// Steady state: ds_load pairs + back-to-back v_wmma with partial dscnt waits.
//
// V_WMMA_F32_16X16X4_F32 operand mapping (wave32):
//  A frag: lane L -> row M=L&15, K = k + 2*(L>=16) + {0,1}
//  B frag: VGPR v -> K = k + 2*(L>=16) + v, N = L&15
//  C frag: VGPR r -> M = r + 8*(L>=16),    N = L&15
// Requires: M % 64 == 0, K % 64 == 0, Nout == 16*NT.
// ---------------------------------------------------------------------------
template <int NT>
__global__ __launch_bounds__(128) void k_gemm_f32(
    const float* __restrict__ A, const float* __restrict__ Bm,
    float* __restrict__ C, int M, int K, int Nout,
    long sA, long sB, long sC, int relu)
{
  constexpr int NOUT = 16 * NT;
  __shared__ float sBuf[64 * NOUT];          // one 64-deep K chunk of B

  const float* Ab = A  + (long)blockIdx.z * sA;
  const float* Bb = Bm + (long)blockIdx.z * sB;   // swizzled
  float*       Cb = C  + (long)blockIdx.z * sC;

  int tid  = threadIdx.x;
  int lane = tid & 31;
  int wave = tid >> 5;
  int n0   = blockIdx.x * NOUT;
  int m0   = (blockIdx.y * WAVES_PER_BLOCK + wave) * 16;

  int mrow  = m0 + (lane & 15);
  int khalf = (lane >> 4) * 2;               // K sub-offset (0 or 2)
  int hi8   = (lane >> 4) * 8;               // C row sub-offset
  int ncol  = lane & 15;

  v8f acc[NT] = {};
  const float* Arow = Ab + (long)mrow * K;

  for (int c = 0; c < K; c += 64) {
    // batch-preload this chunk's A fragments (16 x 8B per lane)
    v2f af[16];
#pragma unroll
    for (int q = 0; q < 16; ++q)
      af[q] = *(const v2f*)(Arow + c + q * 4 + khalf);

    __syncthreads();                         // previous chunk fully consumed
    // cooperative copy: chunk c of swizzled B is contiguous [c*Nout, (c+64)*Nout)
    {
      const float* src = Bb + (long)c * NOUT;
#pragma unroll
      for (int it = 0; it < (64 * NOUT) / (128 * 4); ++it) {
        int idx = tid * 4 + it * 128 * 4;
        *(f32x4*)&sBuf[idx] = *(const f32x4*)&src[idx];
      }
    }
    __syncthreads();

#pragma unroll
    for (int kk = 0; kk < 64; kk += 4) {
      v2f bfr[NT];
#pragma unroll
      for (int t = 0; t < NT; ++t)
        bfr[t] = *(const v2f*)&sBuf[(((kk >> 2) * NOUT) + n0 + 16 * t + ncol) * 4 + khalf];
#pragma unroll
      for (int t = 0; t < NT; ++t)
        acc[t] = __builtin_amdgcn_wmma_f32_16x16x4_f32(
            false, af[kk >> 2], false, bfr[t], (short)0, acc[t], false, false);
    }
  }

#pragma unroll
  for (int t = 0; t < NT; ++t) {
    int col = n0 + 16 * t + ncol;
#pragma unroll
    for (int r = 0; r < 8; ++r) {
      float v = acc[t][r];
      if (relu) v = fmaxf(v, 0.f);
      Cb[(long)(m0 + hi8 + r) * Nout + col] = v;
    }
  }
}

// ---------------------------------------------------------------------------
// Swizzle a row-major B[K,Nout] (batched) into the GEMM's B layout:
//   dst[ ((k>>2)*Nout + n)*4 + (k&3) ] = src[k*Nout + n]
// ---------------------------------------------------------------------------
__global__ void k_swizzleB(const float* __restrict__ src, float* __restrict__ dst,
                           int K, int Nout)
{
  long total = (long)K * Nout;
  long base  = (long)blockIdx.z * total;
  long idx   = (long)blockIdx.x * blockDim.x + threadIdx.x;
  if (idx >= total) return;
  int n = (int)(idx % Nout);
  int k = (int)(idx / Nout);
  dst[base + ((long)(k >> 2) * Nout + n) * 4 + (k & 3)] = src[base + idx];
}

// ---------------------------------------------------------------------------
// text [B,N,D] -> transposed AND swizzled B-operand (K=D rows, Nout=N cols):
//   dst[b*D*N + ((dcol>>2)*N + i)*4 + (dcol&3)] = text[b,i,dcol]
// ---------------------------------------------------------------------------
__global__ void k_transpose_sw(const float* __restrict__ src, float* __restrict__ dst)
{
  const int N = 128, D = 256;
  long idx = (long)blockIdx.x * blockDim.x + threadIdx.x;
  if (idx >= (long)8 * N * D) return;
  int dcol = (int)(idx & (D - 1));
  long t   = idx >> 8;                 // / D
  int i    = (int)(t & (N - 1));
  int b    = (int)(t >> 7);            // / N
  dst[(long)b * D * N + ((long)(dcol >> 2) * N + i) * 4 + (dcol & 3)] = src[idx];
}

// ---------------------------------------------------------------------------
// Masked softmax over j for logits[b,i,:]; one block per (b,i), 128 threads.
// ---------------------------------------------------------------------------
__global__ __launch_bounds__(128) void k_softmax(
    const float* __restrict__ logits, const float* __restrict__ textmask,
    float* __restrict__ probs)
{
  const int N = 128;
  int bi = blockIdx.x;                  // b*N + i
  int b  = bi >> 7;
  int j  = threadIdx.x;
  __shared__ float red[N];

  float x = logits[(long)bi * N + j] + (1.f - textmask[b * N + j]) * -1e20f;
  red[j] = x; __syncthreads();
  for (int s = 64; s > 0; s >>= 1) { if (j < s) red[j] = fmaxf(red[j], red[j + s]); __syncthreads(); }
  float mx = red[0]; __syncthreads();
  float e = __expf(x - mx);
  red[j] = e; __syncthreads();
  for (int s = 64; s > 0; s >>= 1) { if (j < s) red[j] += red[j + s]; __syncthreads(); }
  probs[(long)bi * N + j] = e / red[0];
}

// ---------------------------------------------------------------------------
// outss = outss * textmask[b,i] + align_bias[d]   (in place on d_out half)
// ---------------------------------------------------------------------------
__global__ void k_outss_fin(float* __restrict__ o, const float* __restrict__ mask,
                            const float* __restrict__ abias)
{
  const int N = 128, D = 256;
  long idx = (long)blockIdx.x * blockDim.x + threadIdx.x;
  if (idx >= (long)8 * N * D) return;
  int dcol = (int)(idx & (D - 1));
  long t   = idx >> 8;
  int i    = (int)(t & (N - 1));
  int b    = (int)(t >> 7);
  o[idx] = o[idx] * mask[b * N + i] + abias[dcol];
}

// ---------------------------------------------------------------------------
// Fused gated message passing. One block per (b,i); 8 waves split j; each lane
// owns 8 contiguous d so a wave's loads of each 1KB row are fully coalesced.
// s1/s2 dot products reduced with __shfl_xor (wave32); cross-wave accumulation
// via LDS float atomics; row prefetch (global_prefetch_b8) hides HBM latency.
// ---------------------------------------------------------------------------
union V8 { f32x4 v[2]; float f[8]; };

__device__ __forceinline__ void ld8(V8& r, const float* p) {
  r.v[0] = *(const f32x4*)p;
  r.v[1] = *(const f32x4*)(p + 4);
}

__global__ __launch_bounds__(256) void k_message(
    const float* __restrict__ adj1, const float* __restrict__ adj2,
    const float* __restrict__ e1_fuse, const float* __restrict__ e2_fuse,
    const float* __restrict__ e1_fc,   const float* __restrict__ e2_fc,
    const float* __restrict__ of1, const float* __restrict__ of2,
    const float* __restrict__ pi1, const float* __restrict__ pj1,
    const float* __restrict__ pi2, const float* __restrict__ pj2,
    const float* __restrict__ w1, const float* __restrict__ b1,
    const float* __restrict__ w2, const float* __restrict__ b2,
    float* __restrict__ out1, float* __restrict__ out2)
{
  const int N = 128, D = 256;
  int bi   = blockIdx.x;                // b*N + i
  int b    = bi >> 7;
  int tid  = threadIdx.x;
  int lane = tid & 31;
  int wave = tid >> 5;                  // 0..7
  int d0   = lane * 8;

  __shared__ float aadj1[N], aadj2[N];
  __shared__ float acc1s[D], acc2s[D];
  __shared__ float redn[64];
  __shared__ float dden[2];

  acc1s[tid] = 0.f; acc2s[tid] = 0.f;
  if (tid < N) {
    aadj1[tid] = adj1[(long)bi * N + tid];
    aadj2[tid] = adj2[(long)bi * N + tid];
  }
  __syncthreads();
  if (tid < 64) redn[tid] = aadj1[tid] + aadj1[tid + 64];
  __syncthreads();
  for (int s = 32; s > 0; s >>= 1) { if (tid < s) redn[tid] += redn[tid + s]; __syncthreads(); }
  if (tid == 0) dden[0] = redn[0] + 1.f;
  __syncthreads();
  if (tid < 64) redn[tid] = aadj2[tid] + aadj2[tid + 64];
  __syncthreads();
  for (int s = 32; s > 0; s >>= 1) { if (tid < s) redn[tid] += redn[tid + s]; __syncthreads(); }
  if (tid == 0) dden[1] = redn[0] + 1.f;
  __syncthreads();

  V8 pi1v, pi2v, w1v, w2v;
  ld8(pi1v, pi1 + (long)bi * D + d0);
  ld8(pi2v, pi2 + (long)bi * D + d0);
  ld8(w1v,  w1 + d0);
  ld8(w2v,  w2 + d0);
  float bb1 = b1[0], bb2 = b2[0];

  float a1[8] = {0,0,0,0,0,0,0,0};
  float a2[8] = {0,0,0,0,0,0,0,0};

  long baseBI = (long)bi * N;           // e-tensor row base (b,i,*)
  long baseBJ = (long)b * N;            // per-j activation base (b,*)

  for (int j = wave; j < N; j += 8) {
    long erow = (baseBI + j) * (long)D + d0;
    long jrow = (baseBJ + j) * (long)D + d0;

    if (j + 8 < N) {
      long nrow = (baseBI + j + 8) * (long)D + d0;
      __builtin_prefetch(e1_fc + nrow, 0, 0);
      __builtin_prefetch(e2_fc + nrow, 0, 0);
      __builtin_prefetch(e1_fuse + nrow, 0, 0);
      __builtin_prefetch(e2_fuse + nrow, 0, 0);
    }

    V8 ef1, ef2, pj1v, pj2v;
    ld8(ef1, e1_fc + erow);
    ld8(ef2, e2_fc + erow);
    ld8(pj1v, pj1 + jrow);
    ld8(pj2v, pj2 + jrow);

    float dot1 = 0.f, dot2 = 0.f;
#pragma unroll
    for (int e = 0; e < 8; ++e) {
      float r1 = fmaxf(pi1v.f[e] + pj1v.f[e] + ef1.f[e], 0.f);
      float r2 = fmaxf(pi2v.f[e] + pj2v.f[e] + ef2.f[e], 0.f);
      dot1 += r1 * w1v.f[e];
      dot2 += r2 * w2v.f[e];
    }
#pragma unroll
    for (int off = 16; off > 0; off >>= 1) {
      dot1 += __shfl_xor(dot1, off, 32);
      dot2 += __shfl_xor(dot2, off, 32);
    }
    float s1 = 1.f / (1.f + __expf(-(dot1 + bb1)));
    float s2 = 1.f / (1.f + __expf(-(dot2 + bb2)));
    float wg1 = aadj1[j] * s1;
    float wg2 = aadj2[j] * s2;

    V8 fu1, fu2, o1v, o2v;
    ld8(fu1, e1_fuse + erow);
    ld8(fu2, e2_fuse + erow);
    ld8(o1v, of1 + jrow);
    ld8(o2v, of2 + jrow);
#pragma unroll
    for (int e = 0; e < 8; ++e) {
      a1[e] += wg1 * fmaxf(o1v.f[e] + fu1.f[e], 0.f);
      a2[e] += wg2 * fmaxf(o2v.f[e] + fu2.f[e], 0.f);
    }
  }

#pragma unroll
  for (int e = 0; e < 8; ++e) {
    atomicAdd(&acc1s[d0 + e], a1[e]);
    atomicAdd(&acc2s[d0 + e], a2[e]);
  }
  __syncthreads();
  out1[(long)bi * D + tid] = acc1s[tid] / dden[0];
  out2[(long)bi * D + tid] = acc2s[tid] / dden[1];
}

// ---------------------------------------------------------------------------
// dst = LayerNorm( relu(g1+g2) + cur + bias ) * gamma + beta ; block per (b,i)
// ---------------------------------------------------------------------------
__global__ __launch_bounds__(256) void k_ln(
    const float* __restrict__ g1, const float* __restrict__ g2,
    const float* __restrict__ cur, const float* __restrict__ bias,
    const float* __restrict__ gamma, const float* __restrict__ beta,
    float* __restrict__ dst)
{
  const int D = 256;
  int bi = blockIdx.x;
  int d  = threadIdx.x;
  long idx = (long)bi * D + d;
  float v = fmaxf(g1[idx] + g2[idx], 0.f) + cur[idx] + bias[d];
  __shared__ float red[D];
  red[d] = v; __syncthreads();
  for (int s = 128; s > 0; s >>= 1) { if (d < s) red[d] += red[d + s]; __syncthreads(); }
  float mean = red[0] * (1.f / D);
  __syncthreads();
  float c = v - mean;
  red[d] = c * c; __syncthreads();
  for (int s = 128; s > 0; s >>= 1) { if (d < s) red[d] += red[d + s]; __syncthreads(); }
  float var = red[0] * (1.f / D);
  dst[idx] = c * rsqrtf(var + 1e-5f) * gamma[d] + beta[d];
}

// ---------------------------------------------------------------------------
extern "C" void kernel_launch(void* const* d_in, const int* in_sizes, int n_in,
                              void* d_out, int out_size, void* d_ws, size_t ws_size,
                              hipStream_t stream)
{
  (void)in_sizes; (void)n_in; (void)out_size; (void)ws_size;
  const int B = 8, N = 128, D = 256;
  const long BN = (long)B * N, BND = BN * D, BNN = BN * N, BNND = BNN * D;
  const long DD = (long)D * D;

  const float* text     = (const float*)d_in[0];
  const float* adj1     = (const float*)d_in[1];
  const float* adj2     = (const float*)d_in[2];
  const float* edge1    = (const float*)d_in[3];
  const float* edge2    = (const float*)d_in[4];
  const float* textmask = (const float*)d_in[5];
  const float* weight   = (const float*)d_in[6];
  const float* bias     = (const float*)d_in[7];
  const float* gamma    = (const float*)d_in[8];
  const float* beta     = (const float*)d_in[9];
  const float* fuse1_w  = (const float*)d_in[10];
  const float* fuse2_w  = (const float*)d_in[11];
  const float* fc3_w    = (const float*)d_in[12];
  const float* fc1_w0   = (const float*)d_in[13];
  const float* fc1_w1   = (const float*)d_in[14];
  const float* fc1_b1   = (const float*)d_in[15];
  const float* fc2_w0   = (const float*)d_in[16];
  const float* fc2_w1   = (const float*)d_in[17];
  const float* fc2_b1   = (const float*)d_in[18];
  const float* align_lin  = (const float*)d_in[19];
  const float* align_bias = (const float*)d_in[20];

  float* out_final = (float*)d_out;        // [B,N,D]
  float* outss     = (float*)d_out + BND;  // [B,N,D]

  float* ws = (float*)d_ws;
  size_t off = 0;
  auto alloc = [&](long n) {
    float* p = ws + off;
    off += ((size_t)n + 63) & ~(size_t)63;
    return p;
  };
  float* e1_fuse = alloc(BNND);
  float* e2_fuse = alloc(BNND);
  float* e1_fc   = alloc(BNND);
  float* e2_fc   = alloc(BNND);
  float* sw_textT = alloc((long)B * D * N);   // text^T, swizzled (logits B)
  float* sw_textB = alloc((long)B * N * D);   // text as B of probs GEMM, swizzled
  float* ta      = alloc(BND);
  float* logits  = alloc(BNN);
  float* probs   = alloc(BNN);
  float* cur     = alloc(BND);
  float* of1 = alloc(BND); float* of2 = alloc(BND);
  float* pi1 = alloc(BND); float* pj1 = alloc(BND);
  float* pi2 = alloc(BND); float* pj2 = alloc(BND);
  float* o1  = alloc(BND); float* o2  = alloc(BND);
  float* g1  = alloc(BND); float* g2  = alloc(BND);
  // swizzled DxD weight matrices
  float* sw_align  = alloc(DD);
  float* sw_weight = alloc(DD);
  float* sw_f1a = alloc(DD); float* sw_f1b = alloc(DD);
  float* sw_f2a = alloc(DD); float* sw_f2b = alloc(DD);
  float* sw_c1a = alloc(DD); float* sw_c1b = alloc(DD); float* sw_c1c = alloc(DD);
  float* sw_c2a = alloc(DD); float* sw_c2b = alloc(DD); float* sw_c2c = alloc(DD);
  float* sw_c3a = alloc(DD); float* sw_c3b = alloc(DD);

  dim3 blk(128);
  auto gemm = [&](const float* A, const float* Bsw, float* C, long M, int K, int Nout,
                  long sA, long sB, long sC, int batch, int relu) {
    dim3 grd(1, (unsigned)(M / 64), (unsigned)batch);
    if (Nout == 256)
      k_gemm_f32<16><<<grd, blk, 0, stream>>>(A, Bsw, C, (int)M, K, Nout, sA, sB, sC, relu);
    else
      k_gemm_f32<8><<<grd, blk, 0, stream>>>(A, Bsw, C, (int)M, K, Nout, sA, sB, sC, relu);
  };
  auto swz = [&](const float* src, float* dst, int K, int Nout, int batch) {
    long total = (long)K * Nout;
    dim3 grd((unsigned)((total + 255) / 256), 1, (unsigned)batch);
    k_swizzleB<<<grd, dim3(256), 0, stream>>>(src, dst, K, Nout);
  };

  // ---- weight / operand swizzle prep (tiny) ----
  swz(align_lin,        sw_align,  D, D, 1);
  swz(weight,           sw_weight, D, D, 1);
  swz(fuse1_w,          sw_f1a, D, D, 1);  swz(fuse1_w + DD,     sw_f1b, D, D, 1);
  swz(fuse2_w,          sw_f2a, D, D, 1);  swz(fuse2_w + DD,     sw_f2b, D, D, 1);
  swz(fc1_w0,           sw_c1a, D, D, 1);  swz(fc1_w0 + DD,      sw_c1b, D, D, 1);
  swz(fc1_w0 + 2 * DD,  sw_c1c, D, D, 1);
  swz(fc2_w0,           sw_c2a, D, D, 1);  swz(fc2_w0 + DD,      sw_c2b, D, D, 1);
  swz(fc2_w0 + 2 * DD,  sw_c2c, D, D, 1);
  swz(fc3_w,            sw_c3a, D, D, 1);  swz(fc3_w + DD,       sw_c3b, D, D, 1);
  swz(text,             sw_textB, N, D, B);         // per-batch [N,D] B operand
  k_transpose_sw<<<dim3((unsigned)((BND + 255) / 256)), dim3(256), 0, stream>>>(text, sw_textT);

  // ---- SelfAlignment ----
  gemm(text, sw_align, ta, BN, D, D, 0, 0, 0, 1, 0);                    // ta = text @ align_lin
  gemm(ta, sw_textT, logits, N, D, N, (long)N * D, (long)D * N, (long)N * N, B, 0);
  k_softmax<<<dim3((unsigned)BN), dim3(128), 0, stream>>>(logits, textmask, probs);
  gemm(probs, sw_textB, outss, N, N, D, (long)N * N, (long)N * D, (long)N * D, B, 0);
  k_outss_fin<<<dim3((unsigned)((BND + 255) / 256)), dim3(256), 0, stream>>>(outss, textmask, align_bias);

  // ---- GCN init + loop-invariant edge projections (WMMA, A read once) ----
  gemm(text, sw_weight, cur, BN, D, D, 0, 0, 0, 1, 1);                  // relu(text@weight)
  gemm(edge1, sw_f1b, e1_fuse, BNN, D, D, 0, 0, 0, 1, 0);
  gemm(edge2, sw_f2b, e2_fuse, BNN, D, D, 0, 0, 0, 1, 0);
  gemm(edge1, sw_c1c, e1_fc,   BNN, D, D, 0, 0, 0, 1, 0);
  gemm(edge2, sw_c2c, e2_fc,   BNN, D, D, 0, 0, 0, 1, 0);

  // ---- K = 3 iterations ----
  for (int it = 0; it < 3; ++it) {
    gemm(cur, sw_f1a, of1, BN, D, D, 0, 0, 0, 1, 0);
    gemm(cur, sw_f2a, of2, BN, D, D, 0, 0, 0, 1, 0);
    gemm(cur, sw_c1a, pi1, BN, D, D, 0, 0, 0, 1, 0);
    gemm(cur, sw_c1b, pj1, BN, D, D, 0, 0, 0, 1, 0);
    gemm(cur, sw_c2a, pi2, BN, D, D, 0, 0, 0, 1, 0);
    gemm(cur, sw_c2b, pj2, BN, D, D, 0, 0, 0, 1, 0);
    k_message<<<dim3((unsigned)BN), dim3(256), 0, stream>>>(
        adj1, adj2, e1_fuse, e2_fuse, e1_fc, e2_fc,
        of1, of2, pi1, pj1, pi2, pj2,
        fc1_w1, fc1_b1, fc2_w1, fc2_b1, o1, o2);
    gemm(o1, sw_c3a, g1, BN, D, D, 0, 0, 0, 1, 0);
    gemm(o2, sw_c3b, g2, BN, D, D, 0, 0, 0, 1, 0);
    k_ln<<<dim3((unsigned)BN), dim3(256), 0, stream>>>(
        g1, g2, cur, bias, gamma, beta, (it == 2) ? out_final : cur);
  }
}